// RNNWWD_83124797047478
// MI455X (gfx1250) — compile-verified
//
#include <hip/hip_runtime.h>

typedef float v2f __attribute__((ext_vector_type(2)));
typedef float v8f __attribute__((ext_vector_type(8)));

#define NODEC    200
#define BATCHC   20
#define HIDDENC  1500
#define NWAVES   13
#define NTHREADS (NWAVES * 32)
#define NKT      50                 // 200 / 4 K-tiles
#define DTF      0.5f
#define SQDT     0.70710678118654752f
#define LN066    (-0.41551544396166582f)   // ln(1 - RHO) = ln(0.66)

// shared memory layout (floats):
//   ALs    [NWAVES*NKT*32*2]  packed WMMA A tiles       (41600 floats)
//   Els    [208]              E state vector
//   Yls    [208]              matvec result
//   rowsum [208]              row sums of w_n
//   red    [NTHREADS]         reduction scratch
#define SM_ALS   0
#define SM_ELS   (NWAVES * NKT * 64)          // 41600
#define SM_YLS   (SM_ELS + 208)
#define SM_ROW   (SM_YLS + 208)
#define SM_RED   (SM_ROW + 208)
#define SM_TOTAL (SM_RED + NTHREADS)          // 42640 floats = 170560 bytes

__device__ __forceinline__ float h_tf(float a, float b, float d, float z) {
    float az  = a * z - b;
    float num = 1e-5f + fabsf(az);
    float den = 1e-5f * d + fabsf(1.0f - expf(-d * az));
    return num / den;
}

__device__ __forceinline__ float wl_ij(const float* __restrict__ sc,
                                       const float* __restrict__ wbb,
                                       int i, int j) {
    float wij = expf(wbb[i * NODEC + j]) * sc[i * NODEC + j];
    float wji = expf(wbb[j * NODEC + i]) * sc[j * NODEC + i];
    return log1pf(0.5f * (wij + wji));
}

__device__ __forceinline__ float pow3125(float v) {   // v^(1/0.32) = v^3.125
    float r = sqrtf(sqrtf(sqrtf(v)));                 // v^0.125
    return v * v * v * r;
}

__global__ __launch_bounds__(NTHREADS, 1)
void RNNWWD_kernel(const float* __restrict__ ext,   // (NODE,HIDDEN,BATCH,6)
                   const float* __restrict__ hx,    // (NODE,6)
                   const float* __restrict__ sc,    // (NODE,NODE)
                   const float* __restrict__ wbb,   // (NODE,NODE)
                   const float* __restrict__ g_,
                   const float* __restrict__ gEE_,
                   const float* __restrict__ gIE_,
                   const float* __restrict__ gEI_,
                   const float* __restrict__ sdi_,
                   float* __restrict__ out)         // (NODE,BATCH)
{
    extern __shared__ float smem[];
    float* ALs    = smem + SM_ALS;
    float* Els    = smem + SM_ELS;
    float* Yls    = smem + SM_YLS;
    float* rowsum = smem + SM_ROW;
    float* red    = smem + SM_RED;

    const int  tid     = threadIdx.x;
    const int  wave    = tid >> 5;
    const int  lane    = tid & 31;
    const bool lo      = (lane < 16);
    const int  rowbase = wave * 16;
    const int  row     = rowbase + (lane & 15);
    const int  koff    = lo ? 0 : 2;

    // ---------- preprocessing: 1/Frobenius-norm of w_l ----------
    float partial = 0.0f;
    for (int idx = tid; idx < NODEC * NODEC; idx += NTHREADS) {
        int i = idx / NODEC, j = idx % NODEC;
        float wl = wl_ij(sc, wbb, i, j);
        partial += wl * wl;
    }
    red[tid] = partial;
    __syncthreads();
    if (tid == 0) {
        float s = 0.0f;
        for (int k = 0; k < NTHREADS; ++k) s += red[k];
        red[0] = 1.0f / sqrtf(s);
    }
    __syncthreads();
    const float inv_norm = red[0];

    if (tid < NODEC) {                       // row sums of w_n
        float s = 0.0f;
        for (int j = 0; j < NODEC; ++j) s += wl_ij(sc, wbb, tid, j);
        rowsum[tid] = s * inv_norm;
    }
    __syncthreads();

    // ---------- stage packed A tiles of l_s into LDS ----------
    // V_WMMA_F32_16X16X4_F32 A layout: lanes 0-15 hold (K=k0,k0+1),
    // lanes 16-31 hold (K=k0+2,k0+3), rows M = rowbase + (lane&15).
    for (int kt = 0; kt < NKT; ++kt) {       // heavy body: NOT unrolled, fine
        int k0 = kt * 4 + koff;
        float a0 = 0.0f, a1 = 0.0f;
        if (row < NODEC) {
            a0 = wl_ij(sc, wbb, row, k0)     * inv_norm;
            a1 = wl_ij(sc, wbb, row, k0 + 1) * inv_norm;
            if (row == k0)     a0 -= rowsum[row];
            if (row == k0 + 1) a1 -= rowsum[row];
        }
        int idx = ((wave * NKT + kt) * 32 + lane) * 2;
        ALs[idx]     = a0;
        ALs[idx + 1] = a1;
    }
    __syncthreads();

    // ---------- pull A tiles into VGPRs (trivial body -> full unroll) ----
    v2f Ar[NKT];
    {
        const v2f* ap = (const v2f*)(ALs + wave * NKT * 64) + lane;
#pragma unroll
        for (int kt = 0; kt < NKT; ++kt)
            Ar[kt] = ap[kt * 32];
    }

    // ---------- initial carry ----------
    float E = 0.f, I = 0.f, x = 0.f, f = 1.f, v = 1.f, q = 1.f;
    if (tid < NODEC) {
        E = hx[tid * 6 + 0];
        I = hx[tid * 6 + 1];
        x = hx[tid * 6 + 2];
        f = hx[tid * 6 + 3];
        v = hx[tid * 6 + 4];
        q = hx[tid * 6 + 5];
        Els[tid] = E;
    }
    const float g    = g_[0];
    const float gEEp = 0.001f + fmaxf(gEE_[0], 0.0f);
    const float gIEp = 0.001f + fmaxf(gIE_[0], 0.0f);
    const float gEIp = 0.001f + fmaxf(gEI_[0], 0.0f);
    const float s_in = 0.02f  + fmaxf(sdi_[0], 0.0f);

    // per-node base into the noise tensor; step stride along t is 120 floats,
    // along tr is 6 floats:  idx = nbase + t*120 + tr*6
    const float* nbase = ext + (size_t)tid * (HIDDENC * BATCHC * 6);

    // noise for step 0 (software pipeline prologue)
    float n0 = 0.0f, n1 = 0.0f;
    if (tid < NODEC) {
        n0 = nbase[0];
        n1 = nbase[1];
    }
    __syncthreads();

    // B operand: per-lane LDS base -> single ds_load_b64 per K-tile.
    const v2f* bp = (const v2f*)(Els + koff);

    // ---------- 20 x 1500 sequential integration steps ----------
    for (int tr = 0; tr < BATCHC; ++tr) {
        for (int t = 0; t < HIDDENC; ++t) {
            // next-step indices without division
            int tn = t + 1, trn = tr;
            if (tn == HIDDENC) { tn = 0; trn = tr + 1; }

            // prefetch NEXT step's noise; latency hides under WMMA chain
            float p0 = 0.0f, p1 = 0.0f;
            if (tid < NODEC && trn < BATCHC) {
                const float* np = nbase + tn * (BATCHC * 6) + trn * 6;
                p0 = np[0];
                p1 = np[1];
            }

            // y = l_s @ E : 50 f32 WMMAs per wave, 2 independent chains
            v8f acc0 = {}, acc1 = {};
#pragma unroll
            for (int kt = 0; kt < NKT; kt += 2) {
                v2f b0 = bp[kt * 2];           // E[4kt+koff .. +1]
                v2f b1 = bp[kt * 2 + 2];       // E[4(kt+1)+koff .. +1]
                acc0 = __builtin_amdgcn_wmma_f32_16x16x4_f32(
                           false, Ar[kt],     false, b0, (short)0, acc0,
                           false, false);
                acc1 = __builtin_amdgcn_wmma_f32_16x16x4_f32(
                           false, Ar[kt + 1], false, b1, (short)0, acc1,
                           false, false);
            }
            v8f acc = acc0 + acc1;

            // extract column N=0: lane0 -> rows 0..7, lane16 -> rows 8..15
            if (lane == 0 || lane == 16) {
                int yb = rowbase + (lo ? 0 : 8);
                float4 y0 = {acc[0], acc[1], acc[2], acc[3]};
                float4 y1 = {acc[4], acc[5], acc[6], acc[7]};
                *(float4*)(Yls + yb)     = y0;
                *(float4*)(Yls + yb + 4) = y1;
            }
            __syncthreads();

            if (tid < NODEC) {
                float y  = Yls[tid];
                float IE = fmaxf(0.32f + gEEp * E + g * y - gIEp * I, 0.0f);
                float II = fmaxf(0.7f * 0.32f + gEIp * E - I, 0.0f);
                float rE = h_tf(310.0f, 125.0f, 0.16f,  IE);
                float rI = h_tf(615.0f, 177.0f, 0.087f, II);
                float En = E + DTF * (-E / 100.0f + 0.000641f * (1.0f - E) * rE)
                             + SQDT * n0 * s_in;
                float In = I + DTF * (-I / 10.0f + 0.001f * rI)
                             + SQDT * n1 * s_in;
                E = tanhf(1e-5f + fmaxf(En, 0.0f));
                I = tanhf(1e-5f + fmaxf(In, 0.0f));
                x = x + DTF * (E - x / 0.65f - (f - 1.0f) / 0.41f);
                f = f + DTF * x;
                float vp = pow3125(v);                    // old v
                v = v + DTF * (f - vp) / 0.98f;
                float vpn = pow3125(v);                   // new v
                q = q + DTF * (f * (1.0f - expf(LN066 / f)) / 0.34f
                               - q * vpn / v) / 0.98f;
                Els[tid] = E;
            }
            n0 = p0;
            n1 = p1;
            __syncthreads();
        }
        // BOLD readout for this TR
        if (tid < NODEC) {
            float n2   = nbase[(HIDDENC - 1) * (BATCHC * 6) + tr * 6 + 2];
            float bold = 0.02f * n2 + 5.88235294117647f *
                         (2.38f * (1.0f - q) + 2.0f * (1.0f - q / v) +
                          0.48f * (1.0f - v));
            out[tid * BATCHC + tr] = bold;
        }
    }
}

extern "C" void kernel_launch(void* const* d_in, const int* in_sizes, int n_in,
                              void* d_out, int out_size, void* d_ws, size_t ws_size,
                              hipStream_t stream) {
    const float* ext = (const float*)d_in[0];  // external
    const float* hx  = (const float*)d_in[1];  // hx
    // d_in[2] = hE (unused by reference)
    const float* sc  = (const float*)d_in[3];
    const float* wbb = (const float*)d_in[4];
    const float* g   = (const float*)d_in[5];
    const float* gEE = (const float*)d_in[6];
    const float* gIE = (const float*)d_in[7];
    const float* gEI = (const float*)d_in[8];
    const float* sdi = (const float*)d_in[9];
    float* out = (float*)d_out;
    (void)in_sizes; (void)n_in; (void)out_size; (void)d_ws; (void)ws_size;

    size_t shbytes = (size_t)SM_TOTAL * sizeof(float);   // 170560 B < 320 KB/WGP
    hipFuncSetAttribute((const void*)RNNWWD_kernel,
                        hipFuncAttributeMaxDynamicSharedMemorySize,
                        (int)shbytes);
    RNNWWD_kernel<<<1, NTHREADS, shbytes, stream>>>(ext, hx, sc, wbb,
                                                    g, gEE, gIE, gEI, sdi, out);
}